// KNNTorch_18554258719213
// MI455X (gfx1250) — compile-verified
//
#include <hip/hip_runtime.h>

typedef __attribute__((ext_vector_type(2))) float v2f;
typedef __attribute__((ext_vector_type(8))) float v8f;

// Branchless insert of (v, idx) into ascending sorted top-3 {d0<=d1<=d2}.
__device__ __forceinline__ void ins3(float v, int idx,
                                     float& d0, float& d1, float& d2,
                                     int& i0, int& i1, int& i2) {
  bool c2 = v < d2;
  bool c1 = v < d1;
  bool c0 = v < d0;
  d2 = c1 ? d1 : (c2 ? v : d2);
  i2 = c1 ? i1 : (c2 ? idx : i2);
  d1 = c0 ? d0 : (c1 ? v : d1);
  i1 = c0 ? i0 : (c1 ? idx : i1);
  d0 = c0 ? v   : d0;
  i0 = c0 ? idx : i0;
}

// ---------------------------------------------------------------------------
// Pre-pass: pack candidate points into WMMA A-operand order.
// For candidate m (tile t = m/16, row l = m%16), per batch:
//   slot (t*32 + l)      = {x, y}   (A operand for lanes 0-15:  K=0,1)
//   slot (t*32 + 16 + l) = {z, sq}  (A operand for lanes 16-31: K=2,3)
// so the main kernel's A operand is a single aligned b64 load per lane.
// ---------------------------------------------------------------------------
__global__ __launch_bounds__(256) void pack_kernel(
    const float* __restrict__ pts1, float* __restrict__ apack, int N1) {
  const int b = blockIdx.y;
  const int m = blockIdx.x * 256 + threadIdx.x;
  if (m >= N1) return;
  const float* p = pts1 + ((size_t)b * N1 + m) * 3;
  const float x = p[0], y = p[1], z = p[2];
  const float sq = __builtin_fmaf(x, x, __builtin_fmaf(y, y, z * z));
  const int t = m >> 4, l = m & 15;
  float* base = apack + (size_t)b * N1 * 4;
  v2f lo, hi;
  lo.x = x; lo.y = y;
  hi.x = z; hi.y = sq;
  *(v2f*)(base + (size_t)(t * 32 + l) * 2)      = lo;
  *(v2f*)(base + (size_t)(t * 32 + 16 + l) * 2) = hi;
}

// ---------------------------------------------------------------------------
// Main kernel. Two waves (64 threads) per 16-query tile; each wave scans half
// the candidate range, results merged through LDS. Score for query q, cand m:
//   s = ||p1[m]||^2 - 2*p1[m].p2[q]   (drops the per-row-constant ||p2||^2)
// computed 16x16 at a time by V_WMMA_F32_16X16X4_F32 with K=4 operands:
//   A row m  = [x1, y1, z1, sq1]      (pre-packed, one b64 load per lane)
//   B col q  = [-2x2, -2y2, -2z2, 1]
// PACKED=false recomputes A inline (fallback when workspace is too small).
// ---------------------------------------------------------------------------
template <bool PACKED>
__global__ __launch_bounds__(64) void knn_color_kernel(
    const float* __restrict__ pts1, const float* __restrict__ pts2,
    const float* __restrict__ cols1, const float* __restrict__ apack,
    float* __restrict__ out, int N1, int N2) {
  __shared__ float sval[2][16][3];
  __shared__ int   sidx[2][16][3];

  const int tid    = threadIdx.x;
  const int waveId = tid >> 5;        // 0 or 1: which candidate half
  const int lane   = tid & 31;
  const int half   = lane >> 4;       // K-half of A/B, row-half of D
  const int l16    = lane & 15;
  const int nQT    = N2 >> 4;
  const int b      = blockIdx.x / nQT;
  const int qt     = blockIdx.x - b * nQT;
  const int q      = (qt << 4) + l16;

  const float* p1 = pts1  + (size_t)b * N1 * 3;
  const float* p2 = pts2  + (size_t)b * N2 * 3;
  const float* c1 = cols1 + (size_t)b * N1 * 3;

  // B operand: column for query q (loaded once per wave)
  const float qx = p2[q * 3 + 0];
  const float qy = p2[q * 3 + 1];
  const float qz = p2[q * 3 + 2];
  v2f bmat;
  bmat.x = half ? (-2.0f * qz) : (-2.0f * qx);
  bmat.y = half ? 1.0f         : (-2.0f * qy);

  float d0 = 3.0e38f, d1 = 3.0e38f, d2 = 3.0e38f;
  int   i0 = 0, i1 = 0, i2 = 0;

  const int nT   = N1 >> 4;           // total candidate tiles
  const int nT2  = nT >> 1;           // tiles per wave
  const int tBeg = waveId * nT2;

  // Process one candidate tile: WMMA + cheap reject + (rare) insertion.
  auto process = [&](int tAbs, v2f amat) {
    v8f cacc = {};
    v8f dres = __builtin_amdgcn_wmma_f32_16x16x4_f32(
        false, amat, false, bmat, (short)0, cacc, false, false);

    const float m01 = fminf(dres[0], dres[1]);
    const float m23 = fminf(dres[2], dres[3]);
    const float m45 = fminf(dres[4], dres[5]);
    const float m67 = fminf(dres[6], dres[7]);
    const float mn  = fminf(fminf(m01, m23), fminf(m45, m67));

    // Expected ~3*ln(2048) ~ 23 updates/lane over the whole scan, so this
    // wave-uniform branch skips the 80-op block for almost every tile.
    if (__builtin_amdgcn_ballot_w32(mn < d2) != 0u) {
      const int mbase = (tAbs << 4) + (half << 3);
#pragma unroll
      for (int v = 0; v < 8; ++v) {
        ins3(dres[v], mbase + v, d0, d1, d2, i0, i1, i2);
      }
    }
  };

  if (PACKED) {
    // A operand pre-packed in lane order: one aligned b64 load per tile.
    const float* ap = apack + (size_t)b * N1 * 4 +
                      (size_t)(tBeg * 32 + lane) * 2;
    v2f a = *(const v2f*)ap;
#pragma unroll 2
    for (int t = 0; t < nT2 - 1; ++t) {
      const v2f an = *(const v2f*)(ap + 64);   // next tile: 32 slots * 2
      process(tBeg + t, a);
      ap += 64; a = an;
    }
    process(tBeg + nT2 - 1, a);
  } else {
    // Fallback: build A inline from raw points.
    auto makeA = [&](float x, float y, float z) {
      const float sq = __builtin_fmaf(x, x, __builtin_fmaf(y, y, z * z));
      v2f a;
      a.x = half ? z  : x;
      a.y = half ? sq : y;
      return a;
    };
    const float* ap = p1 + (size_t)(tBeg * 16 + l16) * 3;
    float x = ap[0], y = ap[1], z = ap[2];
#pragma unroll 2
    for (int t = 0; t < nT2 - 1; ++t) {
      const float* np = ap + 48;
      const float nx = np[0], ny = np[1], nz = np[2];
      process(tBeg + t, makeA(x, y, z));
      ap = np; x = nx; y = ny; z = nz;
    }
    process(tBeg + nT2 - 1, makeA(x, y, z));
  }

  // Lanes L and L+16 each saw half the candidate rows for query q: merge.
  {
    float pd0 = __shfl_xor(d0, 16, 32);
    float pd1 = __shfl_xor(d1, 16, 32);
    float pd2 = __shfl_xor(d2, 16, 32);
    int   pi0 = __shfl_xor(i0, 16, 32);
    int   pi1 = __shfl_xor(i1, 16, 32);
    int   pi2 = __shfl_xor(i2, 16, 32);
    ins3(pd0, pi0, d0, d1, d2, i0, i1, i2);
    ins3(pd1, pi1, d0, d1, d2, i0, i1, i2);
    ins3(pd2, pi2, d0, d1, d2, i0, i1, i2);
  }

  // Cross-wave merge through LDS (each wave covered half the candidates).
  if (half == 0) {
    sval[waveId][l16][0] = d0; sidx[waveId][l16][0] = i0;
    sval[waveId][l16][1] = d1; sidx[waveId][l16][1] = i1;
    sval[waveId][l16][2] = d2; sidx[waveId][l16][2] = i2;
  }
  __syncthreads();

  if (waveId == 0 && half == 0) {
#pragma unroll
    for (int j = 0; j < 3; ++j) {
      ins3(sval[1][l16][j], sidx[1][l16][j], d0, d1, d2, i0, i1, i2);
    }

    // Gather the 3 nearest colors and average.
    float r = 0.0f, g = 0.0f, bl = 0.0f;
    const int idxs[3] = {i0, i1, i2};
#pragma unroll
    for (int j = 0; j < 3; ++j) {
      const float* cc = c1 + (size_t)idxs[j] * 3;
      r  += cc[0];
      g  += cc[1];
      bl += cc[2];
    }
    const float s = 1.0f / 3.0f;
    float* o = out + ((size_t)b * N2 + q) * 3;
    o[0] = r * s;
    o[1] = g * s;
    o[2] = bl * s;
  }
}

extern "C" void kernel_launch(void* const* d_in, const int* in_sizes, int n_in,
                              void* d_out, int out_size, void* d_ws, size_t ws_size,
                              hipStream_t stream) {
  (void)n_in; (void)out_size;
  const float* pts1  = (const float*)d_in[0];  // [B, N1, 3]
  const float* pts2  = (const float*)d_in[1];  // [B, N2, 3]
  const float* cols1 = (const float*)d_in[2];  // [B, N1, 3]
  float* out = (float*)d_out;                  // [B, N2, 3]

  const int B  = 2;
  const int N1 = in_sizes[0] / (B * 3);
  const int N2 = in_sizes[1] / (B * 3);

  const size_t packBytes = (size_t)B * N1 * 4 * sizeof(float);
  const int    blocks    = B * (N2 / 16);      // 2 waves per 16-query tile

  if (d_ws != nullptr && ws_size >= packBytes) {
    float* apack = (float*)d_ws;
    dim3 pg((N1 + 255) / 256, B);
    pack_kernel<<<pg, 256, 0, stream>>>(pts1, apack, N1);
    knn_color_kernel<true><<<blocks, 64, 0, stream>>>(
        pts1, pts2, cols1, apack, out, N1, N2);
  } else {
    knn_color_kernel<false><<<blocks, 64, 0, stream>>>(
        pts1, pts2, cols1, nullptr, out, N1, N2);
  }
}